// HypergraphNN_80255758893061
// MI455X (gfx1250) — compile-verified
//
#include <hip/hip_runtime.h>

typedef __attribute__((ext_vector_type(2))) float v2f;
typedef __attribute__((ext_vector_type(8))) float v8f;

#define N_NODES 20000
#define N_EDGES 8000
#define NF 64
#define EPSF 1e-10f
#define LDS_STRIDE 516   // floats; 516*4 % 16 == 0 (B128-aligned) and (4r+c)%64 conflict-free

#define WMMA_F32(A, B, C) \
  __builtin_amdgcn_wmma_f32_16x16x4_f32(false, (A), false, (B), (short)0, (C), false, false)

// CDNA5 async memory->LDS copy, 16 bytes per active lane (tracked by ASYNCcnt).
__device__ __forceinline__ void async_ld_b128(unsigned lds_addr, const void* gptr) {
  asm volatile("global_load_async_to_lds_b128 %0, %1, off"
               :: "v"(lds_addr), "v"(gptr)
               : "memory");
}
__device__ __forceinline__ void wait_asynccnt0() {
  asm volatile("s_wait_asynccnt 0x0" ::: "memory");
}

// ---------------------------------------------------------------------------
// Kernel A: one wave per node row. dv_inv_sqrt[row] = rsqrt(sum_e H[row,e]+eps)
// and X1[row,:] = X[row,:] * dv_inv_sqrt[row]. One streaming read of H.
// ---------------------------------------------------------------------------
__global__ __launch_bounds__(256) void degree_scale_kernel(
    const float* __restrict__ H, const float* __restrict__ X,
    float* __restrict__ dvis, float* __restrict__ X1)
{
  const int wave = threadIdx.x >> 5;
  const int lane = threadIdx.x & 31;
  const int row  = blockIdx.x * 8 + wave;

  const float4* Hrow = reinterpret_cast<const float4*>(H + (size_t)row * N_EDGES);
  float acc = 0.f;
  for (int i = lane; i < N_EDGES / 4; i += 32) {
    float4 v = Hrow[i];
    acc += v.x + v.y + v.z + v.w;
  }
  for (int m = 16; m >= 1; m >>= 1) acc += __shfl_xor(acc, m, 32);

  const float s = rsqrtf(acc + EPSF);
  if (lane == 0) dvis[row] = s;

  X1[row * NF + lane]      = X[row * NF + lane]      * s;
  X1[row * NF + lane + 32] = X[row * NF + lane + 32] * s;
}

// ---------------------------------------------------------------------------
// GEMM1: Y2 = D_e^{-1} (H^T X1).  [E x 64], K = N = 20000.
// 16 waves per 16-wide e-tile, round-robin K-blocks of 4, all four f-tiles per
// wave. A-fragments H[k, e0+m] are coalesced 64B rows; WGP-scope SW prefetch
// leads the 2MB-strided HBM stream. d_e accumulated for free from the
// A-fragments. LDS tree reduction -> deterministic stores (no fp atomics).
// ---------------------------------------------------------------------------
__global__ __launch_bounds__(512) void gemm1_kernel(
    const float* __restrict__ H, const float* __restrict__ X1,
    float* __restrict__ Y2)
{
  __shared__ float red[8 * 1024];
  __shared__ float de_part[16 * 32];
  __shared__ float de_inv_s[16];

  const int tid  = threadIdx.x;
  const int wv   = tid >> 5;
  const int lane = tid & 31;
  const int e0   = blockIdx.x * 16;
  const int m    = lane & 15;
  const int half = lane >> 4;   // 0: K={0,1}, 1: K={2,3}
  const int cc   = half * 2;

  const int wv_u = __builtin_amdgcn_readfirstlane(wv);  // uniform -> scalar control flow

  v8f acc[4] = {v8f{}, v8f{}, v8f{}, v8f{}};
  float dsum = 0.f;

  for (int kb = wv_u; kb < N_NODES / 4; kb += 16) {
    const int krow = kb * 4 + cc;
    const float a0 = H[krow * N_EDGES + e0 + m];
    const float a1 = H[(krow + 1) * N_EDGES + e0 + m];

    // lead the 2MB-strided HBM stream 2 iterations (128 rows) ahead; WGP scope
    if (kb + 32 < N_NODES / 4) {
      __builtin_prefetch(H + (krow + 128) * N_EDGES + e0 + m, 0, 3);
      __builtin_prefetch(H + (krow + 129) * N_EDGES + e0 + m, 0, 3);
    }

    dsum += a0 + a1;
    const v2f A = {a0, a1};

    const float* xb = X1 + krow * NF + m;   // rows krow, krow+1 (L2 resident)
#pragma unroll
    for (int ft = 0; ft < 4; ++ft) {
      const v2f B = {xb[ft * 16], xb[NF + ft * 16]};
      acc[ft] = WMMA_F32(A, B, acc[ft]);
    }
  }

  de_part[wv_u * 32 + lane] = dsum;

  for (int s = 8; s >= 1; s >>= 1) {
    if (wv_u >= s && wv_u < 2 * s) {          // scalar branch, no exec juggling
      float* dst = red + (wv_u - s) * 1024;
#pragma unroll
      for (int ft = 0; ft < 4; ++ft)
#pragma unroll
        for (int i = 0; i < 8; ++i)
          dst[ft * 256 + i * 32 + lane] = acc[ft][i];
    }
    __syncthreads();
    if (wv_u < s) {
      const float* src = red + wv_u * 1024;
#pragma unroll
      for (int ft = 0; ft < 4; ++ft)
#pragma unroll
        for (int i = 0; i < 8; ++i)
          acc[ft][i] += src[ft * 256 + i * 32 + lane];
    }
    __syncthreads();
  }

  if (tid < 16) {
    float s = 0.f;
    for (int w = 0; w < 16; ++w)
      s += de_part[w * 32 + tid] + de_part[w * 32 + 16 + tid];
    de_inv_s[tid] = 1.0f / (s + EPSF);
  }
  __syncthreads();

  if (wv_u == 0) {
#pragma unroll
    for (int ft = 0; ft < 4; ++ft)
#pragma unroll
      for (int i = 0; i < 8; ++i) {
        const int M = i + 8 * half;
        Y2[(e0 + M) * NF + ft * 16 + m] = acc[ft][i] * de_inv_s[M];
      }
  }
}

// ---------------------------------------------------------------------------
// GEMM2: Z = D_v^{-1/2} (H @ Y2).  [N x 64], K = E = 8000.
// 8 waves per 16-wide n-tile. Each K-chunk [16 x 512] of H is staged into LDS
// with GLOBAL_LOAD_ASYNC_TO_LDS_B128 (memory->LDS, no VGPR round trip),
// row stride 516 floats (16B aligned, bank-conflict free). Last chunk covers
// only e<8000: waves 5..7 skip compute there (wave-uniform), so no pad/guards
// are needed around the WMMAs. LDS tree reduction, then scale + store.
// ---------------------------------------------------------------------------
__global__ __launch_bounds__(256) void gemm2_kernel(
    const float* __restrict__ H, const float* __restrict__ Y2,
    const float* __restrict__ dvis, float* __restrict__ Z)
{
  __shared__ float smem[16 * LDS_STRIDE];  // 33 KB tile; reused for reduction (16 KB)

  const int tid  = threadIdx.x;
  const int w    = tid >> 5;
  const int lane = tid & 31;
  const int n0   = blockIdx.x * 16;
  const int m    = lane & 15;
  const int half = lane >> 4;
  const int cc   = half * 2;

  const int w_u = __builtin_amdgcn_readfirstlane(w);

  // staging map: 16 threads per row, one float4 each, 8 passes -> 512 cols
  const int sr = tid >> 4;
  const int cb = (tid & 15) * 4;
  const unsigned lds_row = (unsigned)(size_t)(&smem[sr * LDS_STRIDE]);

  v8f acc[4] = {v8f{}, v8f{}, v8f{}, v8f{}};

  for (int chunk = 0; chunk < 16; ++chunk) {
    const int ebase = chunk * 512;
#pragma unroll
    for (int j = 0; j < 8; ++j) {
      const int c = cb + j * 64;
      const int e = ebase + c;
      if (e < N_EDGES)  // per-lane EXEC mask; OOB lanes transfer nothing
        async_ld_b128(lds_row + (unsigned)c * 4u,
                      H + (size_t)(n0 + sr) * N_EDGES + e);
    }
    wait_asynccnt0();
    __syncthreads();

    // last chunk: waves 5..7 lie entirely in e>=8000 -> skip (uniform branch)
    const int kk_limit = (chunk < 15 || w_u < 5) ? 64 : 0;
    const int eoff = w_u * 64;
    for (int kk = 0; kk < kk_limit; kk += 4) {
      const int base = m * LDS_STRIDE + eoff + kk + cc;
      const v2f A = {smem[base], smem[base + 1]};
      const int erow = ebase + eoff + kk + cc;          // < 8000 always here
      const float* yb = Y2 + erow * NF + m;             // L2 resident
#pragma unroll
      for (int ft = 0; ft < 4; ++ft) {
        const v2f B = {yb[ft * 16], yb[NF + ft * 16]};
        acc[ft] = WMMA_F32(A, B, acc[ft]);
      }
    }
    __syncthreads();
  }

  // tree-reduce accumulators across the 8 waves (reuse smem)
  for (int s = 4; s >= 1; s >>= 1) {
    if (w_u >= s && w_u < 2 * s) {            // scalar branch
      float* dst = smem + (w_u - s) * 1024;
#pragma unroll
      for (int ft = 0; ft < 4; ++ft)
#pragma unroll
        for (int i = 0; i < 8; ++i)
          dst[ft * 256 + i * 32 + lane] = acc[ft][i];
    }
    __syncthreads();
    if (w_u < s) {
      const float* src = smem + w_u * 1024;
#pragma unroll
      for (int ft = 0; ft < 4; ++ft)
#pragma unroll
        for (int i = 0; i < 8; ++i)
          acc[ft][i] += src[ft * 256 + i * 32 + lane];
    }
    __syncthreads();
  }

  if (w_u == 0) {
#pragma unroll
    for (int ft = 0; ft < 4; ++ft)
#pragma unroll
      for (int i = 0; i < 8; ++i) {
        const int M = i + 8 * half;
        const int row = n0 + M;
        Z[row * NF + ft * 16 + m] = acc[ft][i] * dvis[row];
      }
  }
}

// ---------------------------------------------------------------------------
extern "C" void kernel_launch(void* const* d_in, const int* in_sizes, int n_in,
                              void* d_out, int out_size, void* d_ws, size_t ws_size,
                              hipStream_t stream) {
  const float* H = (const float*)d_in[0];   // [20000, 8000] fp32
  const float* X = (const float*)d_in[1];   // [20000, 64]   fp32
  float* Z = (float*)d_out;                 // [20000, 64]   fp32

  float* ws   = (float*)d_ws;
  float* dvis = ws;                                   // 20000 floats
  float* X1   = ws + N_NODES;                         // 20000*64 floats
  float* Y2   = ws + N_NODES + N_NODES * NF;          // 8000*64 floats

  degree_scale_kernel<<<N_NODES / 8, 256, 0, stream>>>(H, X, dvis, X1);
  gemm1_kernel<<<N_EDGES / 16, 512, 0, stream>>>(H, X1, Y2);
  gemm2_kernel<<<N_NODES / 16, 256, 0, stream>>>(H, Y2, dvis, Z);
}